// modified_GraphUNet_3513283248167
// MI455X (gfx1250) — compile-verified
//
#include <hip/hip_runtime.h>
#include <hip/hip_bf16.h>
#include <math.h>

typedef float v2f __attribute__((ext_vector_type(2)));
typedef float v8f __attribute__((ext_vector_type(8)));

// Global-address-space pointer types: force global_load/global_store (no flat)
typedef const float __attribute__((address_space(1)))* cgp;
typedef float       __attribute__((address_space(1)))* gp;
typedef const v2f   __attribute__((address_space(1)))* cgp2;

#define C_FEAT 128
#define N0 4096
#define N1 2048
#define N2 1024

// ---------------------------------------------------------------------------
// fp32 WMMA GEMM:  D[M x N] = op(A)[M x K] * B[K x N] (+bias) (+relu)
//   TRANSA=0: op(A)[m,k] = A[m*lda + k]
//   TRANSA=1: op(A)[m,k] = A[k*lda + m]
//   NFULL=1 : N is a multiple of 16 (no column bounds anywhere)
// One wave computes a 64x16 strip = 4 stacked 16x16 WMMA tiles sharing one
// B fragment (4 independent accumulators pipeline the matrix unit).
// M must be a multiple of 64; K a multiple of 4.
// Addressing: uniform SGPR bases + one 32-bit per-lane element index per
// operand (saddr-form global loads); constant subtile offsets fold into the
// instruction IOFFSET. Only two v_add_nc_u32 per K-step remain.
// ---------------------------------------------------------------------------
template <int TRANSA, int NFULL>
__global__ void gemm_wmma_f32_t(const float* __restrict__ A, int lda,
                                const float* __restrict__ B, int ldb,
                                float* __restrict__ D, int ldd,
                                int N, int K,
                                const float* __restrict__ bias, int relu,
                                int totalTiles, int tilesN)
{
  int wave = threadIdx.x >> 5;
  int tile = blockIdx.x * 4 + wave;        // over (M/64) x (N/16)
  if (tile >= totalTiles) return;          // wave-uniform exit (EXEC full for WMMA)
  int tileM = tile / tilesN;
  int tileN = tile - tileM * tilesN;
  int lane = threadIdx.x & 31;
  int half = lane >> 4;                    // selects K pair {0,1} vs {2,3}
  int l    = lane & 15;
  int row0 = tileM * 64;
  int col0 = tileN * 16;

  int  nb  = col0 + l;
  bool nOK = NFULL ? true : (nb < N);
  int  nbc = NFULL ? nb : (nOK ? nb : 0);  // clamped column (branch-free guard)

  v8f acc0 = (v8f){0.f,0.f,0.f,0.f,0.f,0.f,0.f,0.f};
  v8f acc1 = acc0, acc2 = acc0, acc3 = acc0;

  // Uniform SGPR bases; per-lane 32-bit element indices (saddr-form loads).
  cgp aB0, aB1, aB2, aB3, aBhi;
  unsigned idxA, stepA;
  if (!TRANSA) {
    aB0  = (cgp)(uintptr_t)(A + (size_t)(row0 +  0) * lda);
    aB1  = (cgp)(uintptr_t)(A + (size_t)(row0 + 16) * lda);
    aB2  = (cgp)(uintptr_t)(A + (size_t)(row0 + 32) * lda);
    aB3  = (cgp)(uintptr_t)(A + (size_t)(row0 + 48) * lda);
    aBhi = aB0;                                       // unused
    idxA = (unsigned)l * (unsigned)lda + (unsigned)(half << 1);
    stepA = 4u;
  } else {
    aB0  = (cgp)(uintptr_t)A;
    aBhi = (cgp)(uintptr_t)(A + (size_t)lda);         // second K row
    aB1 = aB2 = aB3 = aB0;                            // unused
    idxA = (unsigned)(half << 1) * (unsigned)lda + (unsigned)(row0 + l);
    stepA = 4u * (unsigned)lda;
  }
  cgp bB0 = (cgp)(uintptr_t)B;
  cgp bB1 = (cgp)(uintptr_t)(B + (size_t)ldb);        // second K row
  unsigned idxB  = (unsigned)(half << 1) * (unsigned)ldb + (unsigned)nbc;
  unsigned stepB = 4u * (unsigned)ldb;

  for (int k0 = 0; k0 < K; k0 += 4) {
    v2f bv = { bB0[idxB], bB1[idxB] };
    v2f av0, av1, av2, av3;
    if (!TRANSA) {
      av0 = *(cgp2)&aB0[idxA];
      av1 = *(cgp2)&aB1[idxA];
      av2 = *(cgp2)&aB2[idxA];
      av3 = *(cgp2)&aB3[idxA];
      __builtin_prefetch((const float*)(uintptr_t)&aB0[idxA + 64], 0, 3);
    } else {
      av0 = (v2f){ aB0[idxA +  0], aBhi[idxA +  0] };   // +16/+32/+48 fold
      av1 = (v2f){ aB0[idxA + 16], aBhi[idxA + 16] };   // into IOFFSET
      av2 = (v2f){ aB0[idxA + 32], aBhi[idxA + 32] };
      av3 = (v2f){ aB0[idxA + 48], aBhi[idxA + 48] };
    }
    idxA += stepA;
    idxB += stepB;
    acc0 = __builtin_amdgcn_wmma_f32_16x16x4_f32(false, av0, false, bv, (short)0, acc0, false, false);
    acc1 = __builtin_amdgcn_wmma_f32_16x16x4_f32(false, av1, false, bv, (short)0, acc1, false, false);
    acc2 = __builtin_amdgcn_wmma_f32_16x16x4_f32(false, av2, false, bv, (short)0, acc2, false, false);
    acc3 = __builtin_amdgcn_wmma_f32_16x16x4_f32(false, av3, false, bv, (short)0, acc3, false, false);
  }

  // C/D layout: VGPR r of subtile t -> row (row0 + t*16 + half*8 + r), col = nb
  if (nOK) {
    float bb = bias ? ((cgp)(uintptr_t)bias)[nb] : 0.0f;
    unsigned idxD = (unsigned)(half << 3) * (unsigned)ldd + (unsigned)nb;
    #pragma unroll
    for (int t = 0; t < 4; ++t) {
      v8f a = (t == 0) ? acc0 : (t == 1) ? acc1 : (t == 2) ? acc2 : acc3;
      gp dB = (gp)(uintptr_t)(D + (size_t)(row0 + t * 16) * ldd);
      #pragma unroll
      for (int r = 0; r < 8; ++r) {
        float v = a[r] + bb;
        if (relu) v = fmaxf(v, 0.0f);
        dB[idxD + (unsigned)r * (unsigned)ldd] = v;
      }
    }
  }
}

// ---------------------------------------------------------------------------
// score = tanh((x @ p) / ||p||) ; one wave per row, block-wide norm reduce
// ---------------------------------------------------------------------------
__global__ void score_kernel(const float* __restrict__ x, const float* __restrict__ p,
                             float* __restrict__ out, int n, int c)
{
  __shared__ float red[256];
  float s = 0.f;
  for (int i = threadIdx.x; i < c; i += blockDim.x) s += p[i] * p[i];
  red[threadIdx.x] = s;
  __syncthreads();
  for (int off = 128; off > 0; off >>= 1) {
    if (threadIdx.x < off) red[threadIdx.x] += red[threadIdx.x + off];
    __syncthreads();
  }
  float inv_norm = rsqrtf(red[0]);
  int wave = threadIdx.x >> 5;
  int lane = threadIdx.x & 31;
  int row = blockIdx.x * 8 + wave;
  if (row >= n) return;
  float d = 0.f;
  for (int i = lane; i < c; i += 32) d += x[(size_t)row * c + i] * p[i];
  for (int m = 16; m > 0; m >>= 1) d += __shfl_xor(d, m, 32);
  if (lane == 0) out[row] = tanhf(d * inv_norm);
}

// ---------------------------------------------------------------------------
// Single-workgroup bitonic sort (desc score, asc index tiebreak) in LDS.
// n is a power of two (4096 / 2048). Emits perm/vals for top-k.
// ---------------------------------------------------------------------------
__global__ void topk_sort_kernel(const float* __restrict__ score, int n, int k,
                                 int* __restrict__ perm, float* __restrict__ vals)
{
  __shared__ float sv[4096];
  __shared__ int   si[4096];
  int tid = threadIdx.x, bd = blockDim.x;
  for (int i = tid; i < n; i += bd) { sv[i] = score[i]; si[i] = i; }
  __syncthreads();
  for (int ks = 2; ks <= n; ks <<= 1) {
    for (int j = ks >> 1; j > 0; j >>= 1) {
      for (int i = tid; i < n; i += bd) {
        int ixj = i ^ j;
        if (ixj > i) {
          float v1 = sv[i], v2 = sv[ixj];
          int   a1 = si[i], a2 = si[ixj];
          // "elem ixj should precede elem i" in (score desc, idx asc) order
          bool before_ji = (v2 > v1) || (v2 == v1 && a2 < a1);
          bool dir = ((i & ks) == 0);
          if (dir ? before_ji : !before_ji) {
            sv[i] = v2; sv[ixj] = v1;
            si[i] = a2; si[ixj] = a1;
          }
        }
      }
      __syncthreads();
    }
  }
  for (int i = tid; i < k; i += bd) { perm[i] = si[i]; vals[i] = sv[i]; }
}

// A1 = A with diagonal forced to 1 (remove self loops + add self loops)
__global__ void make_A1_kernel(const float* __restrict__ A, float* __restrict__ A1, int n)
{
  int j = blockIdx.x * blockDim.x + threadIdx.x;
  int i = blockIdx.y;
  if (j < n) A1[(size_t)i * n + j] = (i == j) ? 1.0f : A[(size_t)i * n + j];
}

// out[i,j] = (i==j) ? 0 : A2[perm[i], perm[j]]   (fused zero-diag + double gather)
__global__ void gather_A_kernel(const float* __restrict__ A2, int lda,
                                const int* __restrict__ perm,
                                float* __restrict__ out, int k)
{
  int j = blockIdx.x * blockDim.x + threadIdx.x;
  int i = blockIdx.y;
  if (j < k) out[(size_t)i * k + j] =
      (i == j) ? 0.0f : A2[(size_t)perm[i] * lda + perm[j]];
}

__global__ void copy2d_kernel(const float* __restrict__ src, int lds,
                              float* __restrict__ dst, int ldd, int cols)
{
  int c = blockIdx.x * blockDim.x + threadIdx.x;
  int r = blockIdx.y;
  if (c < cols) dst[(size_t)r * ldd + c] = src[(size_t)r * lds + c];
}

// x_pool[i,:] = x[perm[i],:] * vals[i]
__global__ void gather_scale_kernel(const float* __restrict__ x, const int* __restrict__ perm,
                                    const float* __restrict__ vals, float* __restrict__ out,
                                    int k, int c)
{
  int i = blockIdx.x * blockDim.x + threadIdx.x;
  if (i < k * c) {
    int r = i / c, col = i - r * c;
    out[i] = x[(size_t)perm[r] * c + col] * vals[r];
  }
}

// out[perm[i],:] += x[i,:]   (perm entries distinct -> no atomics needed)
__global__ void scatter_add_kernel(const float* __restrict__ x, const int* __restrict__ perm,
                                   float* __restrict__ out, int k, int c)
{
  int i = blockIdx.x * blockDim.x + threadIdx.x;
  if (i < k * c) {
    int r = i / c, col = i - r * c;
    out[(size_t)perm[r] * c + col] += x[i];
  }
}

// InstanceNorm (affine=False), per-channel stats over nodes; in-place, opt relu
__global__ void inorm_kernel(float* __restrict__ x, int n, int c, int relu)
{
  __shared__ float s1[256], s2[256];
  int ch = blockIdx.x;
  int tid = threadIdx.x;
  float a = 0.f, b = 0.f;
  for (int r = tid; r < n; r += blockDim.x) {
    float v = x[(size_t)r * c + ch];
    a += v; b += v * v;
  }
  s1[tid] = a; s2[tid] = b;
  __syncthreads();
  for (int off = 128; off > 0; off >>= 1) {
    if (tid < off) { s1[tid] += s1[tid + off]; s2[tid] += s2[tid + off]; }
    __syncthreads();
  }
  float mu  = s1[0] / n;
  float var = s2[0] / n - mu * mu;
  float inv = rsqrtf(var + 1e-5f);
  for (int r = tid; r < n; r += blockDim.x) {
    float v = (x[(size_t)r * c + ch] - mu) * inv;
    if (relu) v = fmaxf(v, 0.0f);
    x[(size_t)r * c + ch] = v;
  }
}

// ---------------------------------------------------------------------------
// Host-side helpers
// ---------------------------------------------------------------------------
static inline void gemm(const float* A, int lda, int transA, const float* B, int ldb,
                        float* D, int ldd, int M, int N, int K,
                        const float* bias, int relu, hipStream_t s)
{
  int tilesM = M / 64;                 // M is always a multiple of 64 here
  int tilesN = (N + 15) >> 4;
  int total = tilesM * tilesN;
  int blocks = (total + 3) >> 2;
  bool nfull = (N & 15) == 0;
  if (transA) {
    if (nfull) gemm_wmma_f32_t<1, 1><<<blocks, 128, 0, s>>>(A, lda, B, ldb, D, ldd, N, K, bias, relu, total, tilesN);
    else       gemm_wmma_f32_t<1, 0><<<blocks, 128, 0, s>>>(A, lda, B, ldb, D, ldd, N, K, bias, relu, total, tilesN);
  } else {
    if (nfull) gemm_wmma_f32_t<0, 1><<<blocks, 128, 0, s>>>(A, lda, B, ldb, D, ldd, N, K, bias, relu, total, tilesN);
    else       gemm_wmma_f32_t<0, 0><<<blocks, 128, 0, s>>>(A, lda, B, ldb, D, ldd, N, K, bias, relu, total, tilesN);
  }
}

static inline void copy2d(const float* src, int lds, float* dst, int ldd,
                          int rows, int cols, hipStream_t s)
{
  dim3 g((cols + 255) / 256, rows);
  copy2d_kernel<<<g, 256, 0, s>>>(src, lds, dst, ldd, cols);
}

// TAGConv: out = [x | A^T x | (A^T)^2 x] @ W + b  (concat staged in cbuf, ld=3*cin)
static void tag_conv(const float* Aadj, int n, const float* xin, int cin,
                     const float* W, const float* bias, float* xout,
                     float* cbuf, int relu, hipStream_t s)
{
  int cc = 3 * cin;
  copy2d(xin, cin, cbuf, cc, n, cin, s);                                     // hop 0
  gemm(Aadj, n, 1, xin, cin, cbuf + cin, cc, n, cin, n, nullptr, 0, s);      // hop 1
  gemm(Aadj, n, 1, cbuf + cin, cc, cbuf + 2 * cin, cc, n, cin, n, nullptr, 0, s); // hop 2
  gemm(cbuf, cc, 0, W, C_FEAT, xout, C_FEAT, n, C_FEAT, cc, bias, relu, s);  // linear
}

static void mlp2(const float* x, int n, const float* W1, const float* b1,
                 const float* W2, const float* b2, float* tmp, float* out,
                 int relu_out, hipStream_t s)
{
  gemm(x, C_FEAT, 0, W1, C_FEAT, tmp, C_FEAT, n, C_FEAT, C_FEAT, b1, 1, s);
  gemm(tmp, C_FEAT, 0, W2, C_FEAT, out, C_FEAT, n, C_FEAT, C_FEAT, b2, relu_out, s);
}

extern "C" void kernel_launch(void* const* d_in, const int* in_sizes, int n_in,
                              void* d_out, int out_size, void* d_ws, size_t ws_size,
                              hipStream_t stream)
{
  (void)in_sizes; (void)n_in; (void)out_size; (void)ws_size;
  const float* x_in        = (const float*)d_in[0];
  const float* A           = (const float*)d_in[1];
  const float* conv0_W     = (const float*)d_in[2];
  const float* conv0_b     = (const float*)d_in[3];
  const float* down_conv_W = (const float*)d_in[4];
  const float* down_conv_b = (const float*)d_in[5];
  const float* down_mlp_W1 = (const float*)d_in[6];
  const float* down_mlp_b1 = (const float*)d_in[7];
  const float* down_mlp_W2 = (const float*)d_in[8];
  const float* down_mlp_b2 = (const float*)d_in[9];
  const float* pool_p      = (const float*)d_in[10];
  const float* up_conv_W   = (const float*)d_in[11];
  const float* up_conv_b   = (const float*)d_in[12];
  const float* up_mlp_W1   = (const float*)d_in[13];
  const float* up_mlp_b1   = (const float*)d_in[14];
  const float* up_mlp_W2   = (const float*)d_in[15];
  const float* up_mlp_b2   = (const float*)d_in[16];
  const float* last_W1     = (const float*)d_in[17];
  const float* last_b1     = (const float*)d_in[18];
  const float* last_W2     = (const float*)d_in[19];
  const float* last_b2     = (const float*)d_in[20];
  float* out = (float*)d_out;

  // ---- workspace carve (~150 MB peak) ----
  float* w = (float*)d_ws;
  float* A1buf  = w;  w += (size_t)N0 * N0;       // 64 MB: A1, later pooled A (As[1])
  float* A2buf  = w;  w += (size_t)N0 * N0;       // 64 MB: A2, later level-1 A bufs
  float* xs0    = w;  w += (size_t)N0 * C_FEAT;   // res level 0
  float* xs1    = w;  w += (size_t)N1 * C_FEAT;   // res level 1
  float* xc     = w;  w += (size_t)N0 * C_FEAT;   // current features
  float* tmpA   = w;  w += (size_t)N0 * 3 * C_FEAT; // concat buffer
  float* tmpB   = w;  w += (size_t)N0 * C_FEAT;
  float* upbuf  = w;  w += (size_t)N0 * C_FEAT;
  float* scores = w;  w += N0;
  float* vals0  = w;  w += N1;
  float* vals1  = w;  w += N2;
  int* perm0    = (int*)w;  w += N1;
  int* perm1    = (int*)w;  w += N2;

  // ---- conv0: x = relu(tag_conv(A, x_in)) ----
  tag_conv(A, N0, x_in, 16, conv0_W, conv0_b, xs0, tmpA, 1, stream);

  // ---- down level 0 ----
  { dim3 g((N0 + 255) / 256, N0); make_A1_kernel<<<g, 256, 0, stream>>>(A, A1buf, N0); }
  gemm(A1buf, N0, 0, A1buf, N0, A2buf, N0, N0, N0, N0, nullptr, 0, stream);  // 4096^3 WMMA
  score_kernel<<<(N0 + 7) / 8, 256, 0, stream>>>(xs0, pool_p, scores, N0, C_FEAT);
  topk_sort_kernel<<<1, 1024, 0, stream>>>(scores, N0, N1, perm0, vals0);
  gather_scale_kernel<<<(N1 * C_FEAT + 255) / 256, 256, 0, stream>>>(xs0, perm0, vals0, xc, N1, C_FEAT);
  { dim3 g((N1 + 255) / 256, N1); gather_A_kernel<<<g, 256, 0, stream>>>(A2buf, N0, perm0, A1buf, N1); }
  // A1buf now = pooled adjacency As[1] (2048x2048)
  tag_conv(A1buf, N1, xc, C_FEAT, down_conv_W, down_conv_b, tmpB, tmpA, 0, stream);
  mlp2(tmpB, N1, down_mlp_W1, down_mlp_b1, down_mlp_W2, down_mlp_b2, xc, xs1, 1, stream);

  // ---- down level 1 (reuse A2buf: [A1s | A2s | Ap2]) ----
  float* A1s = A2buf;
  float* A2s = A2buf + (size_t)N1 * N1;
  float* Ap2 = A2buf + 2 * (size_t)N1 * N1;
  { dim3 g((N1 + 255) / 256, N1); make_A1_kernel<<<g, 256, 0, stream>>>(A1buf, A1s, N1); }
  gemm(A1s, N1, 0, A1s, N1, A2s, N1, N1, N1, N1, nullptr, 0, stream);        // 2048^3 WMMA
  score_kernel<<<(N1 + 7) / 8, 256, 0, stream>>>(xs1, pool_p + C_FEAT, scores, N1, C_FEAT);
  topk_sort_kernel<<<1, 1024, 0, stream>>>(scores, N1, N2, perm1, vals1);
  gather_scale_kernel<<<(N2 * C_FEAT + 255) / 256, 256, 0, stream>>>(xs1, perm1, vals1, xc, N2, C_FEAT);
  { dim3 g((N2 + 255) / 256, N2); gather_A_kernel<<<g, 256, 0, stream>>>(A2s, N1, perm1, Ap2, N2); }
  tag_conv(Ap2, N2, xc, C_FEAT, down_conv_W + 3 * C_FEAT * C_FEAT, down_conv_b + C_FEAT,
           tmpB, tmpA, 0, stream);
  mlp2(tmpB, N2, down_mlp_W1 + C_FEAT * C_FEAT, down_mlp_b1 + C_FEAT,
       down_mlp_W2 + C_FEAT * C_FEAT, down_mlp_b2 + C_FEAT, xc, tmpB, 1, stream);
  // current x (1024x128) in tmpB

  // ---- up i=0 (j=1): res=xs1, Aj=A1buf, perm=perm1 ----
  copy2d(xs1, C_FEAT, upbuf, C_FEAT, N1, C_FEAT, stream);
  scatter_add_kernel<<<(N2 * C_FEAT + 255) / 256, 256, 0, stream>>>(tmpB, perm1, upbuf, N2, C_FEAT);
  tag_conv(A1buf, N1, upbuf, C_FEAT, up_conv_W, up_conv_b, xc, tmpA, 0, stream);
  mlp2(xc, N1, up_mlp_W1, up_mlp_b1, up_mlp_W2, up_mlp_b2, tmpB, xc, 0, stream);
  inorm_kernel<<<C_FEAT, 256, 0, stream>>>(xc, N1, C_FEAT, 1);

  // ---- up i=1 (j=0): res=xs0, Aj=A, perm=perm0 ----
  copy2d(xs0, C_FEAT, upbuf, C_FEAT, N0, C_FEAT, stream);
  scatter_add_kernel<<<(N1 * C_FEAT + 255) / 256, 256, 0, stream>>>(xc, perm0, upbuf, N1, C_FEAT);
  tag_conv(A, N0, upbuf, C_FEAT, up_conv_W + 3 * C_FEAT * C_FEAT, up_conv_b + C_FEAT,
           xc, tmpA, 0, stream);
  gemm(xc, C_FEAT, 0, last_W1, C_FEAT, tmpB, C_FEAT, N0, C_FEAT, C_FEAT, last_b1, 1, stream);
  gemm(tmpB, C_FEAT, 0, last_W2, 1, out, 1, N0, 1, C_FEAT, last_b2, 0, stream);
  inorm_kernel<<<1, 256, 0, stream>>>(out, N0, 1, 0);
}